// Decoder_17300128269031
// MI455X (gfx1250) — compile-verified
//
#include <hip/hip_runtime.h>
#include <hip/hip_bf16.h>
#include <math.h>

// Problem constants (fixed by the reference setup).
#define T_ENC 2048
#define NB    32      // batch
#define DM    256     // d_model
#define HH    512     // hidden
#define NV    32      // vocab
#define NX    512     // 2*d_model (concat [e, ctx])
#define SOS_TOKEN 1
#define NWGS  16      // workgroups in the persistent decode kernel

typedef float v2f __attribute__((ext_vector_type(2)));
typedef float v8f __attribute__((ext_vector_type(8)));

// ---------------------------------------------------------------------------
// K0: enc_score[b,t] = sum_d enc[t,b,d] * W_attn[H + d]   (one-time)
// ---------------------------------------------------------------------------
__global__ void __launch_bounds__(256)
k_scores(const float* __restrict__ enc, const float* __restrict__ W_attn,
         float* __restrict__ scores) {
  const int t   = blockIdx.x;
  const int tid = threadIdx.x;
  const int b   = tid >> 3;
  const int c   = tid & 7;
  const float* p  = enc + (size_t)t * NB * DM + (size_t)b * DM + c * 32;
  const float* we = W_attn + HH + c * 32;
  float s = 0.f;
#pragma unroll
  for (int j = 0; j < 32; ++j) s += p[j] * we[j];
  __shared__ float red[256];
  red[tid] = s;
  __syncthreads();
  if (c == 0) {
    float acc = 0.f;
#pragma unroll
    for (int j = 0; j < 8; ++j) acc += red[b * 8 + j];
    scores[(size_t)b * T_ENC + t] = acc;
  }
}

// ---------------------------------------------------------------------------
// K1: per-b softmax over t, ctx[b,:] = sum_t aw[t]*enc[t,b,:]  (one-time:
// softmax is invariant to the per-b shift (h.W_h + b_attn), so attention
// weights / ctx are step-independent). Also x=[emb[SOS],ctx], h0=0, bar=0.
// ---------------------------------------------------------------------------
__global__ void __launch_bounds__(256)
k_attn_init(const float* __restrict__ enc, const float* __restrict__ scores,
            const float* __restrict__ emb, float* __restrict__ x,
            float* __restrict__ h0, unsigned* __restrict__ bar) {
  const int b   = blockIdx.x;
  const int tid = threadIdx.x;
  __shared__ float p[T_ENC];
  __shared__ float red[256];

  if (b == 0 && tid == 0) *bar = 0u;   // reset persistent-kernel barrier

  float m = -INFINITY;
  for (int t = tid; t < T_ENC; t += 256) m = fmaxf(m, scores[(size_t)b * T_ENC + t]);
  red[tid] = m;
  __syncthreads();
  for (int s = 128; s > 0; s >>= 1) {
    if (tid < s) red[tid] = fmaxf(red[tid], red[tid + s]);
    __syncthreads();
  }
  m = red[0];
  __syncthreads();

  float sum = 0.f;
  for (int t = tid; t < T_ENC; t += 256) {
    float e = expf(scores[(size_t)b * T_ENC + t] - m);
    p[t] = e;
    sum += e;
  }
  red[tid] = sum;
  __syncthreads();
  for (int s = 128; s > 0; s >>= 1) {
    if (tid < s) red[tid] += red[tid + s];
    __syncthreads();
  }
  const float inv = 1.0f / red[0];
  __syncthreads();

  float acc = 0.f;
  for (int t = 0; t < T_ENC; ++t)
    acc += p[t] * enc[(size_t)t * NB * DM + (size_t)b * DM + tid];
  x[b * NX + DM + tid] = acc * inv;                  // ctx half of x
  x[b * NX + tid]      = emb[SOS_TOKEN * DM + tid];  // e = emb[SOS]
  h0[b * HH + tid]       = 0.f;
  h0[b * HH + 256 + tid] = 0.f;
}

// ---------------------------------------------------------------------------
// Grid-wide barrier for the persistent kernel (all NWGS blocks co-resident).
// Monotonic counter: after the n-th barrier, *bar == n*NWGS.
// ---------------------------------------------------------------------------
__device__ __forceinline__ void grid_sync(unsigned* bar, unsigned goal) {
  __syncthreads();
  if (threadIdx.x == 0) {
    __threadfence();  // release this WG's global writes to agent scope
    __hip_atomic_fetch_add(bar, 1u, __ATOMIC_RELEASE, __HIP_MEMORY_SCOPE_AGENT);
    while (__hip_atomic_load(bar, __ATOMIC_ACQUIRE, __HIP_MEMORY_SCOPE_AGENT) < goal) {
      __builtin_amdgcn_s_sleep(1);
    }
    __threadfence();  // acquire: invalidate WGP caches before re-reading
  }
  __syncthreads();
}

// ---------------------------------------------------------------------------
// Persistent decode kernel: all L steps in one launch.
//   Phase A (16 WGs): GRU step. WG wg owns h-columns wg*32..wg*32+31.
//     12 waves = 3 gates x 2 Mtiles x 2 Ntiles; each wave accumulates BOTH
//     the gi tile (A=x, B=W_ih) and gh tile (A=h, B=W_hh) with
//     V_WMMA_F32_16X16X4_F32 over K=512; gates meet in LDS.
//   Phase B (WG 0): pred = h_new @ W_out^T + b_out (WMMA, waves 0-3),
//     write out[b,s,:], argmax -> tok, x[:,0:DM] = emb[tok].
// x, h, out are cross-WG shared within the launch: no __restrict__.
// ---------------------------------------------------------------------------
__global__ void __launch_bounds__(384)
k_decode(float* x, float* h,
         const float* __restrict__ W_ih, const float* __restrict__ W_hh,
         const float* __restrict__ b_ih, const float* __restrict__ b_hh,
         const float* __restrict__ W_out, const float* __restrict__ b_out,
         const float* __restrict__ emb, float* out, int L, unsigned* bar) {
  const int wg   = blockIdx.x;      // 0..15
  const int tid  = threadIdx.x;
  const int wave = tid >> 5;        // 0..11
  const int lane = tid & 31;
  const int g    = wave >> 2;       // 0=r 1=z 2=n
  const int mt   = (wave >> 1) & 1;
  const int nt   = wave & 1;
  const int r16  = lane & 15;
  const int koff  = (lane >> 4) * 2;  // A/B frag K offset (ISA f32 layouts)
  const int mbase = (lane >> 4) * 8;  // C/D frag M offset

  __shared__ float s_rz[2][32 * 32];
  __shared__ float s_in[32 * 32];
  __shared__ float s_hn[32 * 32];
  __shared__ float s_pred[32 * 32];
  __shared__ int   stok[32];

  const int wrow = g * HH + wg * 32 + nt * 16 + r16;  // weight row = output col
  const float* Bi = W_ih + (size_t)wrow * NX + koff;
  const float* Bh = W_hh + (size_t)wrow * HH + koff;

  unsigned goal = 0;
  for (int s = 0; s < L; ++s) {
    const float* h_in  = h + (size_t)(s & 1) * NB * HH;
    float*       h_out = h + (size_t)((s + 1) & 1) * NB * HH;

    // ---- Phase A: GRU matmuls + gates --------------------------------
    {
      const float* Ax = x    + (size_t)(mt * 16 + r16) * NX + koff;
      const float* Ah = h_in + (size_t)(mt * 16 + r16) * HH + koff;
      v8f acc_i = {0.f, 0.f, 0.f, 0.f, 0.f, 0.f, 0.f, 0.f};
      v8f acc_h = {0.f, 0.f, 0.f, 0.f, 0.f, 0.f, 0.f, 0.f};
#pragma unroll 4
      for (int kk = 0; kk < NX; kk += 4) {
        v2f ax = *(const v2f*)(Ax + kk);
        v2f ah = *(const v2f*)(Ah + kk);
        v2f bi = *(const v2f*)(Bi + kk);
        v2f bh = *(const v2f*)(Bh + kk);
        acc_i = __builtin_amdgcn_wmma_f32_16x16x4_f32(false, ax, false, bi,
                                                      (short)0, acc_i, false, false);
        acc_h = __builtin_amdgcn_wmma_f32_16x16x4_f32(false, ah, false, bh,
                                                      (short)0, acc_h, false, false);
      }
#pragma unroll
      for (int r = 0; r < 8; ++r) {
        const int idx = (mt * 16 + mbase + r) * 32 + nt * 16 + r16;
        if (g < 2) {
          s_rz[g][idx] = acc_i[r] + acc_h[r];
        } else {
          s_in[idx] = acc_i[r];
          s_hn[idx] = acc_h[r];
        }
      }
      __syncthreads();
      for (int e = tid; e < 32 * 32; e += 384) {
        const int b  = e >> 5;
        const int jj = e & 31;
        const int J  = wg * 32 + jj;
        const float pre_r = s_rz[0][e] + b_ih[J]      + b_hh[J];
        const float pre_z = s_rz[1][e] + b_ih[HH + J] + b_hh[HH + J];
        const float i_n   = s_in[e]    + b_ih[2 * HH + J];
        const float h_n   = s_hn[e]    + b_hh[2 * HH + J];
        const float r = 1.f / (1.f + expf(-pre_r));
        const float z = 1.f / (1.f + expf(-pre_z));
        const float n = tanhf(i_n + r * h_n);
        const float ho = h_in[(size_t)b * HH + J];
        h_out[(size_t)b * HH + J] = (1.f - z) * n + z * ho;
      }
    }
    goal += NWGS;
    grid_sync(bar, goal);   // h_out complete & visible

    // ---- Phase B: output projection + argmax + next embedding (WG 0) --
    if (wg == 0) {
      if (tid < 128) {                 // waves 0..3, fully active
        const int mt2 = wave >> 1, nt2 = wave & 1;
        const float* Ar = h_out + (size_t)(mt2 * 16 + r16) * HH + koff;
        const float* Br = W_out + (size_t)(nt2 * 16 + r16) * HH + koff;
        v8f acc = {0.f, 0.f, 0.f, 0.f, 0.f, 0.f, 0.f, 0.f};
#pragma unroll 4
        for (int kk = 0; kk < HH; kk += 4) {
          v2f a = *(const v2f*)(Ar + kk);
          v2f b = *(const v2f*)(Br + kk);
          acc = __builtin_amdgcn_wmma_f32_16x16x4_f32(false, a, false, b,
                                                      (short)0, acc, false, false);
        }
#pragma unroll
        for (int r = 0; r < 8; ++r)
          s_pred[(mt2 * 16 + mbase + r) * 32 + nt2 * 16 + r16] = acc[r];
      }
      __syncthreads();
      for (int e = tid; e < NB * NV; e += 384) {
        const int b = e >> 5, v = e & 31;
        const float pv = s_pred[e] + b_out[v];
        out[(size_t)b * L * NV + (size_t)s * NV + v] = pv;
        s_pred[e] = pv;
      }
      __syncthreads();
      if (tid < NB) {                  // first-max tie-break == jnp.argmax
        int best = 0;
        float bm = s_pred[tid * 32];
        for (int v = 1; v < NV; ++v) {
          const float pv = s_pred[tid * 32 + v];
          if (pv > bm) { bm = pv; best = v; }
        }
        stok[tid] = best;
      }
      __syncthreads();
      for (int e = tid; e < NB * DM; e += 384) {
        const int b = e >> 8, d = e & 255;
        x[b * NX + d] = emb[(size_t)stok[b] * DM + d];
      }
    }
    goal += NWGS;
    grid_sync(bar, goal);   // x / out for step s complete & visible
  }
}

// ---------------------------------------------------------------------------
extern "C" void kernel_launch(void* const* d_in, const int* in_sizes, int n_in,
                              void* d_out, int out_size, void* d_ws, size_t ws_size,
                              hipStream_t stream) {
  (void)in_sizes; (void)n_in; (void)ws_size;
  const float* enc    = (const float*)d_in[0];
  const float* emb    = (const float*)d_in[1];
  const float* W_attn = (const float*)d_in[2];
  // d_in[3] = b_attn: constant shift inside softmax -> mathematically inert.
  const float* W_ih   = (const float*)d_in[4];
  const float* W_hh   = (const float*)d_in[5];
  const float* b_ih   = (const float*)d_in[6];
  const float* b_hh   = (const float*)d_in[7];
  const float* W_out  = (const float*)d_in[8];
  const float* b_out  = (const float*)d_in[9];
  // d_in[10] = max_length on device; derive host-side from out_size instead.
  const int L = out_size / (NB * NV);

  float*    ws     = (float*)d_ws;
  float*    scores = ws;                    // B*T_ENC
  float*    x      = scores + NB * T_ENC;   // B*NX ([e, ctx])
  float*    h      = x + NB * NX;           // 2 * B*HH (ping-pong)
  unsigned* bar    = (unsigned*)(h + 2 * NB * HH);
  float*    out    = (float*)d_out;

  k_scores<<<T_ENC, 256, 0, stream>>>(enc, W_attn, scores);
  k_attn_init<<<NB, 256, 0, stream>>>(enc, scores, emb, x, h, bar);
  k_decode<<<NWGS, 384, 0, stream>>>(x, h, W_ih, W_hh, b_ih, b_hh,
                                     W_out, b_out, emb, out, L, bar);
}